// GPTJAttention_44951127720218
// MI455X (gfx1250) — compile-verified
//
#include <hip/hip_runtime.h>
#include <math.h>

// ---------------------------------------------------------------------------
// Problem constants (match reference)
// ---------------------------------------------------------------------------
#define S_    16
#define HID_  4096
#define H_    16
#define D_    256
#define BLK_  16
#define MAXB_ 128

typedef __attribute__((ext_vector_type(2))) float v2f;
typedef __attribute__((ext_vector_type(8))) float v8f;

// ---------------------------------------------------------------------------
// M=16 GEMM via V_WMMA_F32_16X16X4_F32, compile-time K/N so every global
// load in the hot loop is base-pointer + immediate offset (no per-load
// 64-bit VALU address math).
//   C[16,N] = A[16,K] * B[K,N]   (row-major fp32)
// One wave owns a 16x16 tile of C and loops K in steps of 4.
// 32-bit A 16x4 fragment: lanes 0-15 -> {k,k+1}, lanes 16-31 -> {k+2,k+3}.
// B 4x16 fragment mirrored in K-halves, lanes = columns.
// B is streamed exactly once (201MB / 67MB); A (256KB) stays L2-resident.
// ---------------------------------------------------------------------------
template <int KDIM, int NDIM>
__global__ void gemm16_wmma_f32(const float* __restrict__ A,
                                const float* __restrict__ B,
                                float* __restrict__ C) {
  const int lane = threadIdx.x & 31;
  const int wave = threadIdx.x >> 5;
  const int n0   = (blockIdx.x * 4 + wave) * 16;

  const int hi  = lane >> 4;    // which K-half of the fragment this lane holds
  const int col = lane & 15;    // M-row for A loads, N-col for B/C

  v8f c = {};
  const float* __restrict__ ap = A + col * KDIM + 2 * hi;        // A[row][2hi]
  const float* __restrict__ bp = B + (2 * hi) * NDIM + n0 + col; // B[2hi][col]

  #pragma unroll 4
  for (int k = 0; k < KDIM; k += 4) {
    v2f a, b;
    a.x = ap[k];            // A[row][k + 2hi]      (immediate offset)
    a.y = ap[k + 1];        // A[row][k + 2hi + 1]
    b.x = bp[0];            // B[k + 2hi][col]      (immediate offset)
    b.y = bp[NDIM];         // B[k + 2hi + 1][col]
    bp += 4 * NDIM;
    // (neg_a, A, neg_b, B, c_mod, C, reuse_a, reuse_b)
    c = __builtin_amdgcn_wmma_f32_16x16x4_f32(false, a, false, b,
                                              (short)0, c, false, false);
  }
  // C/D layout: VGPR i -> M = i + 8*hi, lane -> N = n0 + (lane&15)
  float* __restrict__ cp = C + (8 * hi) * NDIM + n0 + col;
  #pragma unroll
  for (int i = 0; i < 8; ++i) cp[i * NDIM] = c[i];
}

// ---------------------------------------------------------------------------
// Flash-decode paged attention, one block per (seq, head).
// 8 waves stride tokens [0, L-1); each wave computes the full 256-dim q.k
// dot per token (2x float4 per lane = 1KB coalesced), wave32 shfl_xor
// reduce, online softmax, fp32 accumulator. The freshly projected token
// (logical position L-1, the reference's cache scatter) is folded in by its
// owning wave AFTER the clean streaming loop, so the hot loop has no
// per-token branch. Cross-wave combine in LDS.
// ---------------------------------------------------------------------------
#define ATT_WAVES 8

__global__ void paged_attn_decode(const float* __restrict__ qkv,
                                  const float* __restrict__ kc,
                                  const float* __restrict__ vc,
                                  const int* __restrict__ bt,
                                  const int* __restrict__ ctx,
                                  float* __restrict__ attn_out) {
  const int s    = blockIdx.x;
  const int h    = blockIdx.y;
  const int lane = threadIdx.x & 31;
  const int wave = threadIdx.x >> 5;
  const int L    = ctx[s];
  const float scale = 0.0625f;                // D^-0.5 = 1/16

  const float* q = qkv + (long)s * (3 * HID_) + h * D_;
  const float4 q0 = *(const float4*)(q + 4 * lane);
  const float4 q1 = *(const float4*)(q + 128 + 4 * lane);

  float4 a0 = {0.f, 0.f, 0.f, 0.f};
  float4 a1 = {0.f, 0.f, 0.f, 0.f};
  float m = -INFINITY, l = 0.f;

  const int* bts = bt + s * MAXB_;

  // ---- stream cached tokens [0, L-1) ----
  for (int t = wave; t < L - 1; t += ATT_WAVES) {
    const int  blk = bts[t >> 4];
    const long off = ((long)(blk * BLK_ + (t & (BLK_ - 1))) * H_ + h) * D_;
    const float* kp = kc + off;
    const float* vp = vc + off;

    // prefetch next token's K line for this wave (global_prefetch_b8)
    const int tn = t + ATT_WAVES;
    if (tn < L - 1) {
      const int  nb = bts[tn >> 4];
      const long no = ((long)(nb * BLK_ + (tn & (BLK_ - 1))) * H_ + h) * D_;
      __builtin_prefetch(kc + no + 4 * lane, 0, 1);
    }

    const float4 k0 = *(const float4*)(kp + 4 * lane);
    const float4 k1 = *(const float4*)(kp + 128 + 4 * lane);
    float d = q0.x * k0.x + q0.y * k0.y + q0.z * k0.z + q0.w * k0.w
            + q1.x * k1.x + q1.y * k1.y + q1.z * k1.z + q1.w * k1.w;
    #pragma unroll
    for (int o = 16; o > 0; o >>= 1) d += __shfl_xor(d, o, 32);

    const float sc   = d * scale;
    const float m2   = fmaxf(m, sc);
    const float corr = __expf(m - m2);
    const float p    = __expf(sc - m2);
    const float4 v0 = *(const float4*)(vp + 4 * lane);
    const float4 v1 = *(const float4*)(vp + 128 + 4 * lane);
    a0.x = a0.x * corr + p * v0.x;  a0.y = a0.y * corr + p * v0.y;
    a0.z = a0.z * corr + p * v0.z;  a0.w = a0.w * corr + p * v0.w;
    a1.x = a1.x * corr + p * v1.x;  a1.y = a1.y * corr + p * v1.y;
    a1.z = a1.z * corr + p * v1.z;  a1.w = a1.w * corr + p * v1.w;
    l = l * corr + p;
    m = m2;
  }

  // ---- fold in the freshly projected token (position L-1) ----
  if (((L - 1) % ATT_WAVES) == wave) {
    const float* kp = qkv + (long)s * (3 * HID_) + HID_     + h * D_;
    const float* vp = qkv + (long)s * (3 * HID_) + 2 * HID_ + h * D_;
    const float4 k0 = *(const float4*)(kp + 4 * lane);
    const float4 k1 = *(const float4*)(kp + 128 + 4 * lane);
    float d = q0.x * k0.x + q0.y * k0.y + q0.z * k0.z + q0.w * k0.w
            + q1.x * k1.x + q1.y * k1.y + q1.z * k1.z + q1.w * k1.w;
    #pragma unroll
    for (int o = 16; o > 0; o >>= 1) d += __shfl_xor(d, o, 32);
    const float sc   = d * scale;
    const float m2   = fmaxf(m, sc);
    const float corr = __expf(m - m2);
    const float p    = __expf(sc - m2);
    const float4 v0 = *(const float4*)(vp + 4 * lane);
    const float4 v1 = *(const float4*)(vp + 128 + 4 * lane);
    a0.x = a0.x * corr + p * v0.x;  a0.y = a0.y * corr + p * v0.y;
    a0.z = a0.z * corr + p * v0.z;  a0.w = a0.w * corr + p * v0.w;
    a1.x = a1.x * corr + p * v1.x;  a1.y = a1.y * corr + p * v1.y;
    a1.z = a1.z * corr + p * v1.z;  a1.w = a1.w * corr + p * v1.w;
    l = l * corr + p;
    m = m2;
  }

  // ---- cross-wave combine ----
  __shared__ float sm[ATT_WAVES];
  __shared__ float sl[ATT_WAVES];
  __shared__ float sacc[ATT_WAVES][D_];
  if (lane == 0) { sm[wave] = m; sl[wave] = l; }
  *(float4*)&sacc[wave][4 * lane]       = a0;
  *(float4*)&sacc[wave][128 + 4 * lane] = a1;
  __syncthreads();

  const int d = threadIdx.x;            // blockDim.x == 256 == D_
  float gm = -INFINITY;
  #pragma unroll
  for (int w = 0; w < ATT_WAVES; ++w) gm = fmaxf(gm, sm[w]);
  float num = 0.f, den = 0.f;
  #pragma unroll
  for (int w = 0; w < ATT_WAVES; ++w) {
    const float f = __expf(sm[w] - gm);
    num += f * sacc[w][d];
    den += f * sl[w];
  }
  attn_out[(long)s * HID_ + h * D_ + d] = num / den;
}

// ---------------------------------------------------------------------------
// Launch: qkv-proj (WMMA) -> flash decode -> out-proj (WMMA)
// ---------------------------------------------------------------------------
extern "C" void kernel_launch(void* const* d_in, const int* in_sizes, int n_in,
                              void* d_out, int out_size, void* d_ws, size_t ws_size,
                              hipStream_t stream) {
  const float* hidden = (const float*)d_in[0];
  const float* w_qkv  = (const float*)d_in[1];
  const float* w_out  = (const float*)d_in[2];
  const float* kc     = (const float*)d_in[3];
  const float* vc     = (const float*)d_in[4];
  const int*   bt     = (const int*)d_in[5];
  const int*   ctx    = (const int*)d_in[6];
  float*       out    = (float*)d_out;

  float* qkv  = (float*)d_ws;                  // [16, 12288]
  float* attn = qkv + (size_t)S_ * 3 * HID_;   // [16, 4096]

  // QKV projection: [16,4096] x [4096,12288]; 4 waves/block, 16 cols/wave
  gemm16_wmma_f32<HID_, 3 * HID_><<<(3 * HID_) / 64, 128, 0, stream>>>(
      hidden, w_qkv, qkv);
  // Paged attention decode: one block per (seq, head)
  paged_attn_decode<<<dim3(S_, H_), D_, 0, stream>>>(qkv, kc, vc, bt, ctx, attn);
  // Output projection: [16,4096] x [4096,4096]
  gemm16_wmma_f32<HID_, HID_><<<HID_ / 64, 128, 0, stream>>>(attn, w_out, out);
}